// HierarchicalISCOLoss_70454643523704
// MI455X (gfx1250) — compile-verified
//
#include <hip/hip_runtime.h>
#include <math.h>

// Problem constants (fixed by the reference setup).
#define NROWS 65536
#define NCOLS 2000
#define WAVES_PER_BLK 8      // 256 threads = 8 wave32's
#define CHUNK 128            // floats per wave per chunk (32 lanes * 4 floats)
#define NFULL 15             // 15 full chunks = 1920 floats
#define TAIL_LANES 20        // remaining 80 floats = 20 lanes * 4 floats

// ---------------------------------------------------------------------------
// CDNA5 async global->LDS copy (per lane: 16B global -> 16B LDS), ASYNCcnt.
// GV addressing mode: 64-bit VGPR address, saddr = off.
// ---------------------------------------------------------------------------
__device__ __forceinline__ void async_copy_b128(unsigned lds_byte_addr,
                                                const float* gaddr) {
    asm volatile("global_load_async_to_lds_b128 %0, %1, off"
                 :: "v"(lds_byte_addr), "v"(gaddr)
                 : "memory");
}

// Online logsumexp + argmax update, one v_exp_f32 per element.
// Invariant: m = running max, idx = first column achieving m,
//            s = sum_j exp(x_j - m) over elements seen so far.
__device__ __forceinline__ void online_update(float v, int c,
                                              float& m, float& s, int& idx) {
    float t = __expf(-fabsf(v - m));   // exp(-(|v-m|)) == exp(min(v,m)-max(v,m))
    bool gt = v > m;                   // strict > keeps first occurrence
    s   = gt ? __fmaf_rn(s, t, 1.0f) : (s + t);
    idx = gt ? c : idx;
    m   = gt ? v : m;
}

__global__ void __launch_bounds__(256)
hisco_row_kernel(const float* __restrict__ logits,
                 const int*   __restrict__ targets,
                 const float* __restrict__ dist,
                 float*       __restrict__ row_out) {
    __shared__ float smem[WAVES_PER_BLK][2][CHUNK];   // 8 KB: per-wave double buffer

    const int wave = threadIdx.x >> 5;
    const int lane = threadIdx.x & 31;
    const int row  = blockIdx.x * WAVES_PER_BLK + wave;

    const float* rowp = logits + (size_t)row * NCOLS;
    const float* g    = rowp + lane * 4;               // 16B aligned (8000B rows)

    // Low 32 bits of a generic pointer to LDS == workgroup-relative LDS byte offset.
    const unsigned lds0 = (unsigned)(size_t)(&smem[wave][0][0]) + (unsigned)(lane * 16);
    const unsigned lds1 = lds0 + (unsigned)(CHUNK * 4);

    // Pipeline prologue: chunks 0 and 1 in flight (ASYNCcnt = 2).
    async_copy_b128(lds0, g);
    async_copy_b128(lds1, g + CHUNK);

    float m = -INFINITY, s = 0.0f;
    int   idx = 0;

#pragma unroll
    for (int k = 0; k < NFULL; ++k) {
        // Async loads complete in order: cnt<=1 => chunk k landed in LDS.
        if (k == NFULL - 1)
            asm volatile("s_wait_asynccnt 0x0" ::: "memory");
        else
            asm volatile("s_wait_asynccnt 0x1" ::: "memory");

        const float4 v4 = *(const float4*)&smem[wave][k & 1][lane * 4];
        const int c0 = k * CHUNK + lane * 4;
        online_update(v4.x, c0 + 0, m, s, idx);
        online_update(v4.y, c0 + 1, m, s, idx);
        online_update(v4.z, c0 + 2, m, s, idx);
        online_update(v4.w, c0 + 3, m, s, idx);

        if (k < NFULL - 2) {
            // WAR safety: the ds_load above must finish before the async
            // engine refills this buffer.
            asm volatile("s_wait_dscnt 0x0" ::: "memory");
            async_copy_b128((k & 1) ? lds1 : lds0, g + (k + 2) * CHUNK);
        }
    }

    // Tail: columns 1920..1999 (80 floats) via direct 128-bit loads.
    if (lane < TAIL_LANES) {
        const int c0 = NFULL * CHUNK + lane * 4;
        const float4 v4 = *(const float4*)(rowp + c0);
        online_update(v4.x, c0 + 0, m, s, idx);
        online_update(v4.y, c0 + 1, m, s, idx);
        online_update(v4.z, c0 + 2, m, s, idx);
        online_update(v4.w, c0 + 3, m, s, idx);
    }

    // Wave32 butterfly merge of (m, s, idx); ties -> smaller column index.
#pragma unroll
    for (int off = 16; off >= 1; off >>= 1) {
        float om = __shfl_xor(m,   off, 32);
        float os = __shfl_xor(s,   off, 32);
        int   oi = __shfl_xor(idx, off, 32);
        float nm = fmaxf(m, om);
        float ns = __fmaf_rn(s, __expf(m - nm), os * __expf(om - nm));
        bool take = (om > m) || ((om == m) && (oi < idx));
        idx = take ? oi : idx;
        m = nm;
        s = ns;
    }

    if (lane == 0) {
        const int   t  = targets[row];
        const float lt = rowp[t];                        // logits[row, target]
        const float ce = m + __logf(s) - lt;             // logsumexp - x_t
        const float w  = (idx == t) ? 1.0f : dist[t * NCOLS + idx];
        row_out[row] = ce * w;
    }
}

// Deterministic fixed-order reduction of the 65536 per-row values.
__global__ void __launch_bounds__(1024)
hisco_reduce_kernel(const float* __restrict__ row_vals, float* __restrict__ out) {
    __shared__ float sm[1024];
    const int tid = threadIdx.x;
    float s = 0.0f;
    for (int i = tid; i < NROWS; i += 1024)   // 64 strided elements, fixed order
        s += row_vals[i];
    sm[tid] = s;
    __syncthreads();
#pragma unroll
    for (int off = 512; off > 0; off >>= 1) {
        if (tid < off) sm[tid] += sm[tid + off];
        __syncthreads();
    }
    if (tid == 0) out[0] = sm[0] * (1.0f / (float)NROWS);
}

extern "C" void kernel_launch(void* const* d_in, const int* in_sizes, int n_in,
                              void* d_out, int out_size, void* d_ws, size_t ws_size,
                              hipStream_t stream) {
    const float* logits  = (const float*)d_in[0];   // (65536, 2000) fp32
    const int*   targets = (const int*)  d_in[1];   // (65536,) int
    const float* dist    = (const float*)d_in[2];   // (2000, 2000) fp32
    float* row_vals = (float*)d_ws;                 // 65536 * 4B = 256 KB scratch
    float* out      = (float*)d_out;                // scalar fp32

    hisco_row_kernel<<<NROWS / WAVES_PER_BLK, 32 * WAVES_PER_BLK, 0, stream>>>(
        logits, targets, dist, row_vals);
    hisco_reduce_kernel<<<1, 1024, 0, stream>>>(row_vals, out);
}